// Decoder_70446053589528
// MI455X (gfx1250) — compile-verified
//
#include <hip/hip_runtime.h>
#include <hip/hip_bf16.h>

// ---------------------------------------------------------------------------
// CDNA5 (gfx1250) GAT decoder.
// Matrix ops use V_WMMA_F32_16X16X4_F32 (wave32 full-f32 matrix pipe).
// B operands for the big GEMMs are staged in LDS in a K-pair-interleaved
// layout so each lane's WMMA B operand is a single ds_load_b64 (no register
// shuffles).  alpha for the attention GEMM is recomputed on the fly with the
// softmax normalization folded into the exponent (shift = max + log(denom)).
// ---------------------------------------------------------------------------

typedef __attribute__((ext_vector_type(2))) float v2f;
typedef __attribute__((ext_vector_type(8))) float v8f;

static constexpr int N      = 6144;
static constexpr int F_IN   = 1433;
static constexpr int D      = 128;
static constexpr int GAT_IN = 256;
static constexpr int GOUT   = 1433;
static constexpr int HLD    = 1440;   // GOUT padded to multiple of 16
static constexpr int KC     = 64;     // K-chunk staged in LDS
static constexpr int PW     = 200;    // LDS floats per K-pair row (96*2 + pad)
static constexpr int ZLD    = 130;    // LDS row stride for z_u tile (128 cols)
static constexpr float SLOPE = 0.2f;

// D = A(16x4) * B(4x16) + C   -- f32 WMMA
__device__ __forceinline__ v8f wmma_f32(v2f a, v2f b, v8f c) {
    return __builtin_amdgcn_wmma_f32_16x16x4_f32(
        false, a, false, b, (short)0, c, false, false);
}

// ---------------------------------------------------------------------------
// Wp[256][1440] = W zero-padded on columns.
// ---------------------------------------------------------------------------
__global__ __launch_bounds__(256) void wpad_kernel(
    const float* __restrict__ W, float* __restrict__ Wp) {
    const int idx = blockIdx.x * 256 + threadIdx.x;
    if (idx < GAT_IN * HLD) {
        const int r = idx / HLD;
        const int c = idx % HLD;
        Wp[idx] = (c < GOUT) ? W[(size_t)r * GOUT + c] : 0.f;
    }
}

// ---------------------------------------------------------------------------
// feat[:,0:128] = z_u ; feat[:,128:256] = (x / ||x||_1) @ z_a
// ---------------------------------------------------------------------------
__global__ __launch_bounds__(128) void feat_kernel(
    const float* __restrict__ x, const float* __restrict__ z_u,
    const float* __restrict__ z_a, float* __restrict__ feat) {
    __shared__ float xrow[F_IN];
    __shared__ float red[128];
    const int i = blockIdx.x;
    const int t = threadIdx.x;

    float part = 0.f;
    for (int k = t; k < F_IN; k += 128) {
        float v = x[(size_t)i * F_IN + k];
        xrow[k] = v;
        part += fabsf(v);
    }
    red[t] = part;
    __syncthreads();
    for (int s = 64; s > 0; s >>= 1) {
        if (t < s) red[t] += red[t + s];
        __syncthreads();
    }
    const float inv = 1.f / fmaxf(red[0], 1e-12f);

    float acc = 0.f;
    for (int k = 0; k < F_IN; ++k)
        acc = fmaf(xrow[k], z_a[(size_t)k * D + t], acc);

    feat[(size_t)i * GAT_IN + D + t] = acc * inv;
    feat[(size_t)i * GAT_IN + t]     = z_u[(size_t)i * D + t];
}

// ---------------------------------------------------------------------------
// h = feat @ Wp  (M=6144, N=1440, K=256).  Block: 8 row tiles x one 96-col
// group; B chunks (64x96) staged in LDS, K-pair interleaved.
// ---------------------------------------------------------------------------
__global__ __launch_bounds__(256) void h_gemm_kernel(
    const float* __restrict__ feat, const float* __restrict__ Wp,
    float* __restrict__ h) {
    __shared__ float bsh[(KC / 2) * PW];             // 25.6 KB
    const int wid = threadIdx.x >> 5;
    const int grp = blockIdx.x / 15;                 // 48 groups of 8 tiles
    const int tg  = blockIdx.x % 15;
    const int ti  = grp * 8 + wid;
    const int lane = threadIdx.x & 31;
    const int half = lane >> 4;
    const int lm   = lane & 15;
    const int i0 = ti * 16;
    const int n0 = tg * 96;

    const float* ap = feat + (size_t)(i0 + lm) * GAT_IN + half * 2;
    v8f acc[6] = {};

    for (int kc = 0; kc < GAT_IN; kc += KC) {
        __syncthreads();
        for (int idx = threadIdx.x; idx < KC * 96; idx += 256) {
            const int r = idx / 96;
            const int c = idx % 96;
            bsh[(r >> 1) * PW + c * 2 + (r & 1)] =
                Wp[(size_t)(kc + r) * HLD + n0 + c];
        }
        __syncthreads();
        for (int k = 0; k < KC; k += 4) {
            v2f a = *(const v2f*)(ap + kc + k);
            const float* bp = &bsh[((k >> 1) + half) * PW + lm * 2];
#pragma unroll
            for (int c = 0; c < 6; ++c) {
                v2f b = *(const v2f*)(bp + c * 32);   // single ds_load_b64
                acc[c] = wmma_f32(a, b, acc[c]);
            }
        }
    }
#pragma unroll
    for (int c = 0; c < 6; ++c) {
        const int col = n0 + c * 16 + lm;
        float* op = h + (size_t)(i0 + half * 8) * HLD + col;
#pragma unroll
        for (int v = 0; v < 8; ++v) op[(size_t)v * HLD] = acc[c][v];
    }
}

// ---------------------------------------------------------------------------
// el = h @ attn_l, er = h @ attn_r.  One wave32 per row, shuffle reduction.
// ---------------------------------------------------------------------------
__global__ __launch_bounds__(256) void el_er_kernel(
    const float* __restrict__ h, const float* __restrict__ attn_l,
    const float* __restrict__ attn_r, float* __restrict__ el,
    float* __restrict__ er) {
    const int row  = blockIdx.x * 8 + (threadIdx.x >> 5);
    const int lane = threadIdx.x & 31;
    const float* hrow = h + (size_t)row * HLD;
    float sl = 0.f, sr = 0.f;
    for (int j = lane; j < GOUT; j += 32) {
        const float hv = hrow[j];
        sl = fmaf(hv, attn_l[j], sl);
        sr = fmaf(hv, attn_r[j], sr);
    }
    for (int o = 16; o > 0; o >>= 1) {
        sl += __shfl_xor(sl, o, 32);
        sr += __shfl_xor(sr, o, 32);
    }
    if (lane == 0) { el[row] = sl; er[row] = sr; }
}

// ---------------------------------------------------------------------------
// Per-row masked softmax statistics.  Emits shift[i] = rowmax_i + log(sum_i)
// so alpha_ij = mask * exp(leakyrelu(el_i+er_j) - shift_i)  (normalization
// folded into the exponent: one exp + one cndmask per element in the GEMM).
// ---------------------------------------------------------------------------
__global__ __launch_bounds__(256) void softmax_stats_kernel(
    const float* __restrict__ adj, const float* __restrict__ el,
    const float* __restrict__ er, float* __restrict__ shift) {
    __shared__ float sval[N];
    __shared__ float red[256];
    const int i = blockIdx.x;
    const int t = threadIdx.x;
    const float eli = el[i];
    const float* arow = adj + (size_t)i * N;

    float mx = -3.4e38f;
    for (int j = t; j < N; j += 256) {
        const float a = arow[j];
        float tt = eli + er[j];
        tt = tt > 0.f ? tt : SLOPE * tt;
        const float sc = (a > 0.f) ? tt : -1e9f;
        sval[j] = sc;
        mx = fmaxf(mx, sc);
    }
    red[t] = mx;
    __syncthreads();
    for (int s = 128; s > 0; s >>= 1) {
        if (t < s) red[t] = fmaxf(red[t], red[t + s]);
        __syncthreads();
    }
    mx = red[0];
    __syncthreads();

    float sum = 0.f;
    for (int j = t; j < N; j += 256) sum += __expf(sval[j] - mx);
    red[t] = sum;
    __syncthreads();
    for (int s = 128; s > 0; s >>= 1) {
        if (t < s) red[t] += red[t + s];
        __syncthreads();
    }
    if (t == 0) shift[i] = mx + logf(red[0]);
}

// ---------------------------------------------------------------------------
// outputs_u = z_u @ z_u.T  (6144x6144, K=128).
// Block: 8 row tiles x one 64-col group; 64 z_u rows staged once in LDS
// (contiguous K => B operand is already a single ds_load_b64).
// ---------------------------------------------------------------------------
__global__ __launch_bounds__(256) void zu_gemm_kernel(
    const float* __restrict__ z_u, float* __restrict__ out) {
    __shared__ float bsh[64 * ZLD];                  // 33.3 KB
    const int wid = threadIdx.x >> 5;
    const int grp = blockIdx.x / 96;                 // 48 row groups
    const int tg  = blockIdx.x % 96;
    const int ti  = grp * 8 + wid;
    const int lane = threadIdx.x & 31;
    const int half = lane >> 4;
    const int lm   = lane & 15;
    const int i0 = ti * 16;
    const int j0 = tg * 64;

    for (int idx = threadIdx.x; idx < 64 * D; idx += 256) {
        const int r = idx >> 7;
        const int c = idx & 127;
        bsh[r * ZLD + c] = z_u[(size_t)(j0 + r) * D + c];
    }
    __syncthreads();

    const float* ap = z_u + (size_t)(i0 + lm) * D + half * 2;
    v8f acc[4] = {};

    for (int k = 0; k < D; k += 4) {
        v2f a = *(const v2f*)(ap + k);
#pragma unroll
        for (int c = 0; c < 4; ++c) {
            v2f b = *(const v2f*)(&bsh[(c * 16 + lm) * ZLD + half * 2 + k]);
            acc[c] = wmma_f32(a, b, acc[c]);
        }
    }
#pragma unroll
    for (int c = 0; c < 4; ++c) {
        float* op = out + (size_t)(i0 + half * 8) * N + j0 + c * 16 + lm;
#pragma unroll
        for (int v = 0; v < 8; ++v) op[(size_t)v * N] = acc[c][v];
    }
}

// ---------------------------------------------------------------------------
// outputs_a = alpha @ h + bias  (M=6144, N=1440 pad, K=6144).
// alpha = mask * exp(leakyrelu(el_i + er_j) - shift_i), branch-free VALU +
// v_exp_f32.  Block: 8 row tiles x one 96-col group; h chunks (64x96) staged
// in LDS (K-pair interleaved), next chunk prefetched.
// ---------------------------------------------------------------------------
__global__ __launch_bounds__(256) void gat_gemm_kernel(
    const float* __restrict__ adj, const float* __restrict__ h,
    const float* __restrict__ el, const float* __restrict__ er,
    const float* __restrict__ shift, const float* __restrict__ bias,
    float* __restrict__ out) {
    __shared__ float bsh[(KC / 2) * PW];             // 25.6 KB
    const int wid = threadIdx.x >> 5;
    const int grp = blockIdx.x / 15;                 // 48 groups of 8 tiles
    const int tg  = blockIdx.x % 15;
    const int ti  = grp * 8 + wid;
    const int lane = threadIdx.x & 31;
    const int half = lane >> 4;
    const int lm   = lane & 15;
    const int i0 = ti * 16;
    const int n0 = tg * 96;

    const int arow  = i0 + lm;
    const float eli = el[arow];
    const float shi = shift[arow];
    const float* adjp = adj + (size_t)arow * N + half * 2;
    const float* erp  = er + half * 2;

    v8f acc[6] = {};

    for (int kc = 0; kc < N; kc += KC) {
        __syncthreads();
        for (int idx = threadIdx.x; idx < KC * 96; idx += 256) {
            const int r = idx / 96;
            const int c = idx % 96;
            bsh[(r >> 1) * PW + c * 2 + (r & 1)] =
                h[(size_t)(kc + r) * HLD + n0 + c];
        }
        // Prefetch next chunk into cache while this chunk computes.
        if (kc + KC < N) {
            const int r = threadIdx.x >> 2;          // 64 rows, 4 thr/row
            const int c = (threadIdx.x & 3) * 24;
            __builtin_prefetch(h + (size_t)(kc + KC + r) * HLD + n0 + c, 0, 0);
        }
        __syncthreads();

        for (int k = 0; k < KC; k += 4) {
            const v2f ad = *(const v2f*)(adjp + kc + k);
            const v2f ev = *(const v2f*)(erp + kc + k);
            float t0 = eli + ev.x; t0 = t0 > 0.f ? t0 : SLOPE * t0;
            float t1 = eli + ev.y; t1 = t1 > 0.f ? t1 : SLOPE * t1;
            v2f a;
            a.x = (ad.x > 0.f) ? __expf(t0 - shi) : 0.f;
            a.y = (ad.y > 0.f) ? __expf(t1 - shi) : 0.f;

            const float* bp = &bsh[((k >> 1) + half) * PW + lm * 2];
#pragma unroll
            for (int c = 0; c < 6; ++c) {
                v2f b = *(const v2f*)(bp + c * 32);   // single ds_load_b64
                acc[c] = wmma_f32(a, b, acc[c]);
            }
        }
    }
#pragma unroll
    for (int c = 0; c < 6; ++c) {
        const int col = n0 + c * 16 + lm;
        if (col < GOUT) {
            const float bz = bias[col];
            float* op = out + (size_t)(i0 + half * 8) * GOUT + col;
#pragma unroll
            for (int v = 0; v < 8; ++v) op[(size_t)v * GOUT] = acc[c][v] + bz;
        }
    }
}

// ---------------------------------------------------------------------------
extern "C" void kernel_launch(void* const* d_in, const int* in_sizes, int n_in,
                              void* d_out, int out_size, void* d_ws,
                              size_t ws_size, hipStream_t stream) {
    const float* x      = (const float*)d_in[0];
    const float* adj    = (const float*)d_in[1];
    const float* z_u    = (const float*)d_in[2];
    const float* z_a    = (const float*)d_in[3];
    const float* W      = (const float*)d_in[4];
    const float* attn_l = (const float*)d_in[5];
    const float* attn_r = (const float*)d_in[6];
    const float* bias   = (const float*)d_in[7];

    float* out_u = (float*)d_out;                 // [N, N]
    float* out_a = out_u + (size_t)N * N;         // [N, GOUT]

    char* ws = (char*)d_ws;
    float* feat = (float*)(ws);                                   // N*256
    float* hbuf = (float*)(ws + (size_t)N * GAT_IN * 4);          // N*1440
    float* Wp   = (float*)(ws + (size_t)N * (GAT_IN + HLD) * 4);  // 256*1440
    float* el    = Wp + (size_t)GAT_IN * HLD;
    float* er    = el + N;
    float* shift = er + N;

    wpad_kernel<<<(GAT_IN * HLD + 255) / 256, 256, 0, stream>>>(W, Wp);
    feat_kernel<<<N, 128, 0, stream>>>(x, z_u, z_a, feat);
    h_gemm_kernel<<<48 * 15, 256, 0, stream>>>(feat, Wp, hbuf);
    el_er_kernel<<<N / 8, 256, 0, stream>>>(hbuf, attn_l, attn_r, el, er);
    softmax_stats_kernel<<<N, 256, 0, stream>>>(adj, el, er, shift);
    zu_gemm_kernel<<<48 * 96, 256, 0, stream>>>(z_u, out_u);
    gat_gemm_kernel<<<48 * 15, 256, 0, stream>>>(adj, hbuf, el, er, shift,
                                                 bias, out_a);
}